// GAT_layer_86088324481810
// MI455X (gfx1250) — compile-verified
//
#include <hip/hip_runtime.h>
#include <hip/hip_bf16.h>
#include <math.h>

// Problem constants (from reference)
#define N_NODES 100000
#define N_EDGES 1600000
#define IN_DIM  128
#define NH      8
#define NC      16
#define HC      128   // NH*NC
#define EDIM    16
#define LN_EPS  1e-5f

typedef float v2f __attribute__((ext_vector_type(2)));
typedef float v8f __attribute__((ext_vector_type(8)));

// ---- workspace layout (bytes). Total ~115.3 MB ----
static constexpr size_t OFF_V     = 0;                                   // 128 f
static constexpr size_t OFF_X     = 1024;                                // N*128 f
static constexpr size_t OFF_ASRC  = OFF_X    + (size_t)N_NODES*HC*4;     // N*8 f
static constexpr size_t OFF_ADST  = OFF_ASRC + (size_t)N_NODES*NH*4;     // N*8 f
static constexpr size_t OFF_M     = OFF_ADST + (size_t)N_NODES*NH*4;     // N*8 u32 (ordered-encoded)
static constexpr size_t OFF_DEN   = OFF_M    + (size_t)N_NODES*NH*4;     // N*8 f
static constexpr size_t OFF_ALPHA = OFF_DEN  + (size_t)N_NODES*NH*4;     // E*8 f

// monotonic float<->uint encoding so we can use native u32 atomic max
__device__ __forceinline__ unsigned f2ord(float f) {
  unsigned u = __float_as_uint(f);
  return (u & 0x80000000u) ? ~u : (u | 0x80000000u);
}
__device__ __forceinline__ float ord2f(unsigned u) {
  u = (u & 0x80000000u) ? (u & 0x7FFFFFFFu) : ~u;
  return __uint_as_float(u);
}

// K0: zero node accumulator (lives in d_out), init m = enc(-inf), den = 0
__global__ void k_init(float* __restrict__ out_node, unsigned* __restrict__ m,
                       float* __restrict__ den) {
  size_t i = (size_t)blockIdx.x * blockDim.x + threadIdx.x;
  if (i < (size_t)N_NODES * HC) out_node[i] = 0.0f;
  if (i < (size_t)N_NODES * NH) { m[i] = 0x007FFFFFu; /* enc(-inf) */ den[i] = 0.0f; }
}

// K1: fold W_edge with att_edge -> v[k=0..15][h=0..7]
__global__ void k_fold_edge(const float* __restrict__ W_e, const float* __restrict__ att_e,
                            float* __restrict__ v) {
  int i = threadIdx.x;           // 0..127
  int k = i >> 3, h = i & 7;
  float s = 0.0f;
  for (int c = 0; c < NC; ++c) s = fmaf(W_e[k * HC + h * NC + c], att_e[h * NC + c], s);
  v[k * NH + h] = s;
}

// K2: x = point_attr @ W via V_WMMA_F32_16X16X4_F32; fused a_src/a_dst epilogue.
// One wave32 per 16-row tile; N_NODES = 6250 * 16 exactly, so EXEC is all-1s.
__global__ void __launch_bounds__(32)
k_node_gemm(const float* __restrict__ A, const float* __restrict__ W,
            const float* __restrict__ att_s, const float* __restrict__ att_d,
            float* __restrict__ x, float* __restrict__ a_src, float* __restrict__ a_dst) {
  const int lane = threadIdx.x & 31;
  const int half = lane >> 4;     // 0: K+{0,1}, 1: K+{2,3}  (ISA 16x4 f32 A layout)
  const int nn   = lane & 15;
  const int row0 = blockIdx.x * 16;

  // Preload the whole 16x128 A tile as per-lane K-fragments.
  v2f afrag[32];
  const float* ap = A + (size_t)(row0 + nn) * IN_DIM + 2 * half;
#pragma unroll
  for (int k = 0; k < 32; ++k) {
    float2 t = *reinterpret_cast<const float2*>(ap + 4 * k);
    afrag[k][0] = t.x; afrag[k][1] = t.y;
  }

#pragma unroll
  for (int h = 0; h < NH; ++h) {            // output col-tile j == head h (C==16)
    v8f acc = {0.f, 0.f, 0.f, 0.f, 0.f, 0.f, 0.f, 0.f};
#pragma unroll
    for (int k = 0; k < 32; ++k) {
      const float* wp = W + (size_t)(4 * k + 2 * half) * HC + h * NC + nn;
      v2f b; b[0] = wp[0]; b[1] = wp[HC];   // B 4x16: same K->(vgpr,half) striping as A
      acc = __builtin_amdgcn_wmma_f32_16x16x4_f32(false, afrag[k], false, b,
                                                  (short)0, acc, false, false);
    }
    const float as = att_s[h * NC + nn];
    const float ad = att_d[h * NC + nn];
#pragma unroll
    for (int i = 0; i < 8; ++i) {           // C/D layout: vgpr i -> M = i + 8*half
      const int row = row0 + i + 8 * half;
      x[(size_t)row * HC + h * NC + nn] = acc[i];
      float ps = acc[i] * as, pd = acc[i] * ad;
#pragma unroll
      for (int off = 8; off >= 1; off >>= 1) {   // reduce over the 16 N-lanes
        ps += __shfl_xor(ps, off, 16);
        pd += __shfl_xor(pd, off, 16);
      }
      if (nn == 0) {
        a_src[(size_t)row * NH + h] = ps;
        a_dst[(size_t)row * NH + h] = pd;
      }
    }
  }
}

// K3: per-edge attention logits + segment max (ordered-u32 atomic max)
__global__ void __launch_bounds__(256)
k_edge_logits(const int* __restrict__ ei, const float* __restrict__ ea,
              const float* __restrict__ v, const float* __restrict__ a_src,
              const float* __restrict__ a_dst, float* __restrict__ alpha,
              unsigned* __restrict__ m) {
  int e = blockIdx.x * blockDim.x + threadIdx.x;
  if (e >= N_EDGES) return;
  const int s = ei[e], d = ei[N_EDGES + e];
  float f[EDIM];
#pragma unroll
  for (int k = 0; k < EDIM; k += 4) {
    float4 t = *reinterpret_cast<const float4*>(ea + (size_t)e * EDIM + k);
    f[k] = t.x; f[k + 1] = t.y; f[k + 2] = t.z; f[k + 3] = t.w;
  }
  const float* as = a_src + (size_t)s * NH;
  const float* ad = a_dst + (size_t)d * NH;
#pragma unroll
  for (int h = 0; h < NH; ++h) {
    float dot = 0.0f;
#pragma unroll
    for (int k = 0; k < EDIM; ++k) dot = fmaf(f[k], v[k * NH + h], dot);
    float al = as[h] + ad[h] + dot;
    al = al > 0.0f ? al : 0.2f * al;               // leaky_relu 0.2
    alpha[(size_t)e * NH + h] = al;
    atomicMax(&m[(size_t)d * NH + h], f2ord(al));
  }
}

// K4: ex = exp(alpha - m[dst]); accumulate denominator
__global__ void __launch_bounds__(256)
k_edge_exp(const int* __restrict__ ei, float* __restrict__ alpha,
           const unsigned* __restrict__ m, float* __restrict__ den) {
  int e = blockIdx.x * blockDim.x + threadIdx.x;
  if (e >= N_EDGES) return;
  const int d = ei[N_EDGES + e];
#pragma unroll
  for (int h = 0; h < NH; ++h) {
    float mh = ord2f(m[(size_t)d * NH + h]);
    float ex = __expf(alpha[(size_t)e * NH + h] - mh);
    alpha[(size_t)e * NH + h] = ex;
    atomicAdd(&den[(size_t)d * NH + h], ex);
  }
}

// K5: one wave per edge: scatter out[dst] += x[src]*alpha_n, fused edge-LN output
__global__ void __launch_bounds__(256)
k_scatter(const int* __restrict__ ei, const float* __restrict__ x,
          const float* __restrict__ exa, const float* __restrict__ den,
          const float* __restrict__ g, const float* __restrict__ b,
          const float* __restrict__ ew, const float* __restrict__ ebias,
          float* __restrict__ out_node, float* __restrict__ edge_out) {
  const int e    = (int)(((size_t)blockIdx.x * blockDim.x + threadIdx.x) >> 5);
  const int lane = threadIdx.x & 31;
  if (e >= N_EDGES) return;
  const int s = ei[e], d = ei[N_EDGES + e];
  const int h = lane >> 2;                       // cols lane*4..+3 all in head lane/4
  const float an = exa[(size_t)e * NH + h] / (den[(size_t)d * NH + h] + 1e-16f);
  float4 xv = *reinterpret_cast<const float4*>(x + (size_t)s * HC + lane * 4);
  float* op = out_node + (size_t)d * HC + lane * 4;
  atomicAdd(op + 0, xv.x * an);
  atomicAdd(op + 1, xv.y * an);
  atomicAdd(op + 2, xv.z * an);
  atomicAdd(op + 3, xv.w * an);
  if (lane == 0) {                               // edge LayerNorm over 8 heads + relu(dot)
    float a[NH]; float mu = 0.0f;
#pragma unroll
    for (int hh = 0; hh < NH; ++hh) {
      a[hh] = exa[(size_t)e * NH + hh] / (den[(size_t)d * NH + hh] + 1e-16f);
      mu += a[hh];
    }
    mu *= (1.0f / NH);
    float var = 0.0f;
#pragma unroll
    for (int hh = 0; hh < NH; ++hh) { float t = a[hh] - mu; var = fmaf(t, t, var); }
    var *= (1.0f / NH);
    const float rs = rsqrtf(var + LN_EPS);
    float acc = ebias[0];
#pragma unroll
    for (int hh = 0; hh < NH; ++hh)
      acc = fmaf((a[hh] - mu) * rs * g[hh] + b[hh], ew[hh], acc);
    edge_out[e] = fmaxf(acc, 0.0f);
  }
}

// K6: one wave per node: +bias, LayerNorm(128), leaky_relu(0.01), in-place on d_out
__global__ void __launch_bounds__(256)
k_node_finish(float* __restrict__ out_node, const float* __restrict__ bias,
              const float* __restrict__ g, const float* __restrict__ b) {
  const int n    = (int)(((size_t)blockIdx.x * blockDim.x + threadIdx.x) >> 5);
  const int lane = threadIdx.x & 31;
  if (n >= N_NODES) return;
  float* p = out_node + (size_t)n * HC + lane * 4;
  float4 v  = *reinterpret_cast<float4*>(p);
  float4 bv = *reinterpret_cast<const float4*>(bias + lane * 4);
  v.x += bv.x; v.y += bv.y; v.z += bv.z; v.w += bv.w;
  float s = v.x + v.y + v.z + v.w;
#pragma unroll
  for (int off = 16; off >= 1; off >>= 1) s += __shfl_xor(s, off, 32);
  const float mu = s * (1.0f / HC);
  float4 dx = {v.x - mu, v.y - mu, v.z - mu, v.w - mu};
  float q = dx.x * dx.x + dx.y * dx.y + dx.z * dx.z + dx.w * dx.w;
#pragma unroll
  for (int off = 16; off >= 1; off >>= 1) q += __shfl_xor(q, off, 32);
  const float rs = rsqrtf(q * (1.0f / HC) + LN_EPS);
  float4 gv = *reinterpret_cast<const float4*>(g + lane * 4);
  float4 bb = *reinterpret_cast<const float4*>(b + lane * 4);
  float4 r;
  r.x = fmaf(dx.x * rs, gv.x, bb.x); r.y = fmaf(dx.y * rs, gv.y, bb.y);
  r.z = fmaf(dx.z * rs, gv.z, bb.z); r.w = fmaf(dx.w * rs, gv.w, bb.w);
  r.x = r.x > 0.f ? r.x : 0.01f * r.x; r.y = r.y > 0.f ? r.y : 0.01f * r.y;
  r.z = r.z > 0.f ? r.z : 0.01f * r.z; r.w = r.w > 0.f ? r.w : 0.01f * r.w;
  *reinterpret_cast<float4*>(p) = r;
}

extern "C" void kernel_launch(void* const* d_in, const int* in_sizes, int n_in,
                              void* d_out, int out_size, void* d_ws, size_t ws_size,
                              hipStream_t stream) {
  const int*   ei    = (const int*)  d_in[0];   // [2,E]
  const float* pa    = (const float*)d_in[1];   // [N,128]
  const float* ea    = (const float*)d_in[2];   // [E,16]
  const float* W     = (const float*)d_in[3];   // [128,128]
  const float* att_s = (const float*)d_in[4];   // [1,8,16]
  const float* att_d = (const float*)d_in[5];
  const float* W_e   = (const float*)d_in[6];   // [16,128]
  const float* att_e = (const float*)d_in[7];
  const float* bias  = (const float*)d_in[8];   // [128]
  const float* ln_ng = (const float*)d_in[9];
  const float* ln_nb = (const float*)d_in[10];
  const float* ln_eg = (const float*)d_in[11];  // [8]
  const float* ln_eb = (const float*)d_in[12];
  const float* ew    = (const float*)d_in[13];  // [8,1]
  const float* eb    = (const float*)d_in[14];  // [1]

  char* ws = (char*)d_ws;                       // needs ~115.3 MB
  float*    v     = (float*)   (ws + OFF_V);
  float*    x     = (float*)   (ws + OFF_X);
  float*    a_src = (float*)   (ws + OFF_ASRC);
  float*    a_dst = (float*)   (ws + OFF_ADST);
  unsigned* m     = (unsigned*)(ws + OFF_M);
  float*    den   = (float*)   (ws + OFF_DEN);
  float*    alpha = (float*)   (ws + OFF_ALPHA);

  float* out_node = (float*)d_out;
  float* edge_out = out_node + (size_t)N_NODES * HC;

  k_init<<<(N_NODES * HC + 255) / 256, 256, 0, stream>>>(out_node, m, den);
  k_fold_edge<<<1, 128, 0, stream>>>(W_e, att_e, v);
  k_node_gemm<<<N_NODES / 16, 32, 0, stream>>>(pa, W, att_s, att_d, x, a_src, a_dst);
  k_edge_logits<<<(N_EDGES + 255) / 256, 256, 0, stream>>>(ei, ea, v, a_src, a_dst, alpha, m);
  k_edge_exp<<<(N_EDGES + 255) / 256, 256, 0, stream>>>(ei, alpha, m, den);
  k_scatter<<<N_EDGES / 8, 256, 0, stream>>>(ei, x, alpha, den, ln_eg, ln_eb, ew, eb,
                                             out_node, edge_out);
  k_node_finish<<<N_NODES * 32 / 256, 256, 0, stream>>>(out_node, bias, ln_ng, ln_nb);
}